// LargeMarginLoss_7086696038882
// MI455X (gfx1250) — compile-verified
//
#include <hip/hip_runtime.h>
#include <math.h>

// Problem constants from the reference
#define BATCH 128
#define DIM   1024
#define NCLS  1000
#define GAMMA 10.0f
#define EPSI  1e-6f

typedef float v2f __attribute__((ext_vector_type(2)));
typedef float v8f __attribute__((ext_vector_type(8)));

// ---------------------------------------------------------------------------
// GEMM 1: logits[b,c] = sum_d A[b,d] * W[d,c] + bias[c]
// One wave per 16x16 output tile, V_WMMA_F32_16X16X4_F32, K stepped 8/iter
// with two independent accumulator chains.  OOB classes (c >= 1000) are
// handled by CLAMPING the load column: a B-fragment column only feeds its own
// output column, which we never store, so no zero-fill / exec divergence is
// needed inside the WMMA loop (EXEC stays all-1s as the ISA requires).
// grid = (ceil(NCLS/16), BATCH/16), block = 32.
// ---------------------------------------------------------------------------
__global__ void lm_logits_wmma(const float* __restrict__ A,
                               const float* __restrict__ W,
                               const float* __restrict__ bias,
                               float* __restrict__ logits) {
    const int lane = threadIdx.x;        // 0..31
    const int half = lane >> 4;          // 0 or 1 (selects K pair)
    const int l16  = lane & 15;
    const int tm   = blockIdx.y * 16;    // batch-row base
    const int tn   = blockIdx.x * 16;    // class-col base

    const int c  = tn + l16;                       // this lane's output column
    const int cc = c < NCLS ? c : (NCLS - 1);      // clamped (edge-tile safe)

    v8f acc0 = {0.f, 0.f, 0.f, 0.f, 0.f, 0.f, 0.f, 0.f};
    v8f acc1 = acc0;

    const float* arow = A + (size_t)(tm + l16) * DIM;   // row-major A row
    const float* wcol = W + cc;                         // clamped W column

    for (int k0 = 0; k0 < DIM; k0 += 8) {
        const int ka = k0 + 2 * half;
        const int kb = ka + 4;

        v2f a0 = { arow[ka], arow[ka + 1] };
        v2f b0 = { wcol[(size_t)ka * NCLS], wcol[(size_t)(ka + 1) * NCLS] };
        v2f a1 = { arow[kb], arow[kb + 1] };
        v2f b1 = { wcol[(size_t)kb * NCLS], wcol[(size_t)(kb + 1) * NCLS] };

        acc0 = __builtin_amdgcn_wmma_f32_16x16x4_f32(
            false, a0, false, b0, (short)0, acc0, false, false);
        acc1 = __builtin_amdgcn_wmma_f32_16x16x4_f32(
            false, a1, false, b1, (short)0, acc1, false, false);
    }

    if (c < NCLS) {
        const float bi = bias[c];
        #pragma unroll
        for (int v = 0; v < 8; ++v) {
            const int m = tm + v + 8 * half;           // batch row
            logits[(size_t)m * NCLS + c] = acc0[v] + acc1[v] + bi;
        }
    }
}

// ---------------------------------------------------------------------------
// GEMM 2: S[c,b] = sum_d W[d,c] * W[d, targets[b]]   (Gram vs. target cols)
// Output tile: 16 classes (M) x 16 batch entries (N).  A-fragment row m only
// feeds output row m, so the OOB class rows are clamped, not predicated.
// grid = (ceil(NCLS/16), BATCH/16), block = 32.
// ---------------------------------------------------------------------------
__global__ void lm_gram_wmma(const float* __restrict__ W,
                             const int* __restrict__ targets,
                             float* __restrict__ S) {
    const int lane = threadIdx.x;
    const int half = lane >> 4;
    const int l16  = lane & 15;
    const int tc   = blockIdx.x * 16;    // class base (M dim)
    const int tb   = blockIdx.y * 16;    // batch base (N dim)

    const int c  = tc + l16;                       // A-side class for this lane
    const int cc = c < NCLS ? c : (NCLS - 1);      // clamped
    const int tcol = targets[tb + l16];            // B-side target column

    const float* wa = W + cc;      // A fragment column
    const float* wb = W + tcol;    // B fragment column

    v8f acc0 = {0.f, 0.f, 0.f, 0.f, 0.f, 0.f, 0.f, 0.f};
    v8f acc1 = acc0;

    for (int k0 = 0; k0 < DIM; k0 += 8) {
        const int ka = k0 + 2 * half;
        const int kb = ka + 4;
        const size_t ra0 = (size_t)ka * NCLS;
        const size_t ra1 = (size_t)(ka + 1) * NCLS;
        const size_t rb0 = (size_t)kb * NCLS;
        const size_t rb1 = (size_t)(kb + 1) * NCLS;

        v2f a0 = { wa[ra0], wa[ra1] };
        v2f b0 = { wb[ra0], wb[ra1] };
        v2f a1 = { wa[rb0], wa[rb1] };
        v2f b1 = { wb[rb0], wb[rb1] };

        acc0 = __builtin_amdgcn_wmma_f32_16x16x4_f32(
            false, a0, false, b0, (short)0, acc0, false, false);
        acc1 = __builtin_amdgcn_wmma_f32_16x16x4_f32(
            false, a1, false, b1, (short)0, acc1, false, false);
    }

    if (c < NCLS) {
        #pragma unroll
        for (int v = 0; v < 8; ++v) {
            const int bb = tb + v + 8 * half;          // batch index
            S[(size_t)c * BATCH + bb] = acc0[v] + acc1[v];
        }
    }
}

// ---------------------------------------------------------------------------
// colnorm[c] = sum_d W[d,c]^2.  Thread-per-column; adjacent threads read
// adjacent columns -> coalesced 128B lines per iteration.
// ---------------------------------------------------------------------------
__global__ void lm_colnorm(const float* __restrict__ W,
                           float* __restrict__ colnorm) {
    const int c = blockIdx.x * blockDim.x + threadIdx.x;
    if (c >= NCLS) return;
    float s = 0.f;
    for (int d = 0; d < DIM; ++d) {
        const float w = W[(size_t)d * NCLS + c];
        s = fmaf(w, w, s);
    }
    colnorm[c] = s;
}

// ---------------------------------------------------------------------------
// Per-sample loss: one block per batch element b.
//   dn = sqrt(max(||W_c||^2 + ||W_t||^2 - 2 S[c,b], 0) + eps) + eps
//   v  = relu(GAMMA + (logits[b,c] - logits[b,t]) / dn), v[t] = 0
//   lossb[b] = max_c v
// ---------------------------------------------------------------------------
__global__ void lm_rowmax(const float* __restrict__ logits,
                          const float* __restrict__ S,
                          const float* __restrict__ colnorm,
                          const int* __restrict__ targets,
                          float* __restrict__ lossb) {
    __shared__ float red[256];
    const int b = blockIdx.x;
    const int t = targets[b];
    const float correct = logits[(size_t)b * NCLS + t];
    const float tn2 = colnorm[t];

    float vmax = 0.f;   // relu >= 0 and the target entry is zeroed
    for (int c = threadIdx.x; c < NCLS; c += blockDim.x) {
        if (c == t) continue;
        float n2 = colnorm[c] + tn2 - 2.f * S[(size_t)c * BATCH + b];
        n2 = n2 > 0.f ? n2 : 0.f;                 // guard tiny negative roundoff
        const float dn = sqrtf(n2 + EPSI) + EPSI;
        const float sd = logits[(size_t)b * NCLS + c] - correct;
        float v = GAMMA + sd / dn;
        v = v > 0.f ? v : 0.f;
        vmax = fmaxf(vmax, v);
    }

    red[threadIdx.x] = vmax;
    __syncthreads();
    for (int s = blockDim.x >> 1; s > 0; s >>= 1) {
        if (threadIdx.x < s)
            red[threadIdx.x] = fmaxf(red[threadIdx.x], red[threadIdx.x + s]);
        __syncthreads();
    }
    if (threadIdx.x == 0) lossb[b] = red[0];
}

// Final mean over BATCH per-sample losses -> d_out[0] (plain store, no atomics)
__global__ void lm_mean(const float* __restrict__ lossb,
                        float* __restrict__ out) {
    __shared__ float red[BATCH];
    red[threadIdx.x] = lossb[threadIdx.x];
    __syncthreads();
    for (int s = BATCH >> 1; s > 0; s >>= 1) {
        if (threadIdx.x < s) red[threadIdx.x] += red[threadIdx.x + s];
        __syncthreads();
    }
    if (threadIdx.x == 0) out[0] = red[0] * (1.0f / (float)BATCH);
}

// ---------------------------------------------------------------------------
extern "C" void kernel_launch(void* const* d_in, const int* in_sizes, int n_in,
                              void* d_out, int out_size, void* d_ws, size_t ws_size,
                              hipStream_t stream) {
    (void)in_sizes; (void)n_in; (void)out_size; (void)ws_size;

    const float* A    = (const float*)d_in[0];   // [128,1024]
    const float* W    = (const float*)d_in[1];   // [1024,1000]
    const float* bias = (const float*)d_in[2];   // [1000]
    const int*   tgt  = (const int*)d_in[3];     // [128] int32

    // Workspace layout (floats): ~1.03 MB total
    float* ws      = (float*)d_ws;
    float* logits  = ws;                                  // 128*1000
    float* S       = logits + (size_t)BATCH * NCLS;       // 1000*128
    float* colnorm = S + (size_t)NCLS * BATCH;            // 1000
    float* lossb   = colnorm + 1024;                      // 128 (aligned gap)

    const int ntilesC = (NCLS + 15) / 16;   // 63
    const int ntilesB = BATCH / 16;         // 8

    // Producers (stream-ordered)
    lm_logits_wmma<<<dim3(ntilesC, ntilesB), 32, 0, stream>>>(A, W, bias, logits);
    lm_gram_wmma  <<<dim3(ntilesC, ntilesB), 32, 0, stream>>>(W, tgt, S);
    lm_colnorm    <<<dim3((NCLS + 255) / 256), 256, 0, stream>>>(W, colnorm);

    // Consumers
    lm_rowmax<<<BATCH, 256, 0, stream>>>(logits, S, colnorm, tgt, lossb);
    lm_mean  <<<1, BATCH, 0, stream>>>(lossb, (float*)d_out);
}